// PostProcessor_40209483825456
// MI455X (gfx1250) — compile-verified
//
#include <hip/hip_runtime.h>
#include <hip/hip_bf16.h>
#include <math.h>

// Problem constants (match reference)
#define BATCH 2
#define NPROP 1000
#define NCLS  81
#define NDET  100
#define BN    (BATCH * NPROP)         // 2000 rows
#define NPAIR (BATCH * (NCLS - 1))    // 160 independent NMS problems

static constexpr float IMW = 1333.0f, IMH = 800.0f;
static constexpr float SCORE_THR = 0.05f, NMS_THR = 0.5f;
static constexpr float BBOX_CLIP = 4.13516655674235600f; // log(1000/16)

#define NEG_INF (-__builtin_inff())

typedef float v2f __attribute__((ext_vector_type(2)));
typedef float v8f __attribute__((ext_vector_type(8)));

// ---------------------------------------------------------------------------
// Kernel 1: softmax + box decode.
// One wave32 handles a 16-row tile. The softmax denominator is computed as
// rowsum = exp(logits - max) x ones via V_WMMA_F32_16X16X4_F32 (full f32):
//   A (16x4 f32): lanes 0-15 supply K=k0,k0+1 ; lanes 16-31 supply K=k0+2,k0+3
//   B (4x16)    : all ones  ->  D[m,n] = sum_k A[m,k] for every n.
// Row sums are then broadcast out of the D tile with wave32 shuffles
// (lane 0 holds rows 0-7 in VGPRs 0-7, lane 16 holds rows 8-15).
// ---------------------------------------------------------------------------
__global__ __launch_bounds__(256) void softmax_decode_kernel(
    const float* __restrict__ cls, const float* __restrict__ reg,
    const float* __restrict__ prop, float* __restrict__ ws_scores,
    float* __restrict__ ws_boxes)
{
  const int lane = threadIdx.x & 31;
  const int wave = threadIdx.x >> 5;
  const int tile = blockIdx.x * 8 + wave;        // 125 valid tiles of 16 rows
  if (tile * 16 >= BN) return;                   // wave-uniform exit, no barriers used
  const int mrow = lane & 15;
  const int r    = tile * 16 + mrow;             // row handled by this lane (dup in lane pairs)
  const float* row = cls + (size_t)r * NCLS;

  // Row max (each half-wave duplicates its row's scan; L2-hot, trivial traffic)
  float mx = NEG_INF;
  for (int c = 0; c < NCLS; ++c) mx = fmaxf(mx, row[c]);

  // WMMA f32 accumulation of exp(row - max) against a ones B-matrix.
  v8f acc = {};
  v2f bones; bones[0] = 1.0f; bones[1] = 1.0f;
  const int khalf = (lane >= 16) ? 2 : 0;
  for (int k0 = 0; k0 < NCLS; k0 += 4) {         // 21 matrix ops, EXEC full at each
    const int c0 = k0 + khalf, c1 = c0 + 1;
    v2f a;
    a[0] = (c0 < NCLS) ? __expf(row[c0] - mx) : 0.0f;
    a[1] = (c1 < NCLS) ? __expf(row[c1] - mx) : 0.0f;
    acc = __builtin_amdgcn_wmma_f32_16x16x4_f32(false, a, false, bones,
                                                (short)0, acc, false, false);
  }

  // Extract this lane's row sum from the D tile (D[m, 0]).
  float sum = 0.0f;
#pragma unroll
  for (int j = 0; j < 8; ++j) {
    const float lo = __shfl(acc[j], 0, 32);      // rows 0..7
    const float hi = __shfl(acc[j], 16, 32);     // rows 8..15
    sum = (mrow == j)     ? lo : sum;
    sum = (mrow == 8 + j) ? hi : sum;
  }
  const float inv = 1.0f / sum;

  // Proposal geometry for this row.
  const float px1 = prop[r * 4 + 0], py1 = prop[r * 4 + 1];
  const float px2 = prop[r * 4 + 2], py2 = prop[r * 4 + 3];
  const float w  = px2 - px1 + 1.0f, h  = py2 - py1 + 1.0f;
  const float cx = px1 + 0.5f * w,   cy = py1 + 0.5f * h;
  const int   b  = r / NPROP,        n  = r % NPROP;

  // Each half-wave lane writes 40 classes: scores + decoded clipped boxes,
  // laid out class-major per (batch, class) to match flat_s/flat_b ordering.
  const int cstart = (lane < 16) ? 1 : 41;
  for (int c = cstart; c < cstart + 40; ++c) {
    const float sc = __expf(row[c] - mx) * inv;
    const int   p  = (b * (NCLS - 1) + (c - 1)) * NPROP + n;
    ws_scores[p] = sc;

    const float* q  = reg + (size_t)r * (NCLS * 4) + c * 4;
    const float dx = q[0] * 0.1f, dy = q[1] * 0.1f;            // / WX, / WY
    const float dw = fminf(q[2] * 0.2f, BBOX_CLIP);            // / WW, clip
    const float dh = fminf(q[3] * 0.2f, BBOX_CLIP);            // / WH, clip
    const float pcx = dx * w + cx, pcy = dy * h + cy;
    const float pw  = __expf(dw) * w, ph = __expf(dh) * h;
    const float x1 = fminf(fmaxf(pcx - 0.5f * pw,        0.0f), IMW - 1.0f);
    const float y1 = fminf(fmaxf(pcy - 0.5f * ph,        0.0f), IMH - 1.0f);
    const float x2 = fminf(fmaxf(pcx + 0.5f * pw - 1.0f, 0.0f), IMW - 1.0f);
    const float y2 = fminf(fmaxf(pcy + 0.5f * ph - 1.0f, 0.0f), IMH - 1.0f);
    *(float4*)(ws_boxes + (size_t)p * 4) = make_float4(x1, y1, x2, y2);
  }
}

// ---------------------------------------------------------------------------
// Kernel 2: per-(batch,class) NMS. 160 blocks x 256 threads (8 waves).
// LDS bitonic sort of 1024 (key, idx) pairs (descending key, ascending idx on
// ties == JAX stable argsort of -s), then greedy suppression: each of the 1000
// serial steps computes IoU row i against j>i with 256 threads (~4/thread).
// Results are written back in place (regenerated by K1 every launch).
// ---------------------------------------------------------------------------
__global__ __launch_bounds__(256) void nms_kernel(float* __restrict__ ws_scores,
                                                  float* __restrict__ ws_boxes)
{
  const int p   = blockIdx.x;
  const int tid = threadIdx.x;
  float* sc  = ws_scores + (size_t)p * NPROP;
  float* bxg = ws_boxes  + (size_t)p * NPROP * 4;

  __shared__ float key[1024];
  __shared__ int   sidx[1024];
  __shared__ float bx1[1024], by1[1024], bx2[1024], by2[1024], area[1024];
  __shared__ unsigned int keep[1024];

  for (int i = tid; i < 1024; i += 256) {
    const float v = (i < NPROP) ? sc[i] : NEG_INF;
    key[i]  = (v > SCORE_THR) ? v : NEG_INF;
    sidx[i] = i;
  }
  __syncthreads();

  // Bitonic sort, comparator = (key desc, idx asc).
  for (int k = 2; k <= 1024; k <<= 1) {
    for (int j = k >> 1; j > 0; j >>= 1) {
      for (int i = tid; i < 1024; i += 256) {
        const int ixj = i ^ j;
        if (ixj > i) {
          const bool  up = (i & k) == 0;
          const float a = key[i], bk = key[ixj];
          const int   ia = sidx[i], ib = sidx[ixj];
          const bool aFirst = (a > bk) || (a == bk && ia < ib);
          if (up != aFirst) {
            key[i] = bk; key[ixj] = a;
            sidx[i] = ib; sidx[ixj] = ia;
          }
        }
      }
      __syncthreads();
    }
  }

  // Gather boxes in sorted order; precompute areas; init keep = isfinite.
  for (int i = tid; i < 1024; i += 256) {
    const int id = sidx[i];
    const float4 bb = (id < NPROP) ? *(const float4*)(bxg + (size_t)id * 4)
                                   : make_float4(0, 0, 0, 0);
    bx1[i] = bb.x; by1[i] = bb.y; bx2[i] = bb.z; by2[i] = bb.w;
    area[i] = (bb.z - bb.x + 1.0f) * (bb.w - bb.y + 1.0f);
    keep[i] = (key[i] > NEG_INF) ? 1u : 0u;
  }
  __syncthreads();

  // Greedy suppression scan.
  for (int i = 0; i < NPROP; ++i) {
    if (keep[i]) {                                   // uniform across block
      const float x1 = bx1[i], y1 = by1[i], x2 = bx2[i], y2 = by2[i], ai = area[i];
      for (int j = i + 1 + tid; j < NPROP; j += 256) {
        if (keep[j]) {
          float iw = fminf(x2, bx2[j]) - fmaxf(x1, bx1[j]) + 1.0f;
          float ih = fminf(y2, by2[j]) - fmaxf(y1, by1[j]) + 1.0f;
          iw = fmaxf(iw, 0.0f); ih = fmaxf(ih, 0.0f);
          const float inter = iw * ih;
          const float iou = inter / (ai + area[j] - inter);
          if (iou > NMS_THR) keep[j] = 0u;
        }
      }
    }
    __syncthreads();
  }

  // Write sorted boxes and suppressed scores back in place.
  for (int i = tid; i < NPROP; i += 256) {
    sc[i] = keep[i] ? key[i] : NEG_INF;
    *(float4*)(bxg + (size_t)i * 4) = make_float4(bx1[i], by1[i], bx2[i], by2[i]);
  }
}

// ---------------------------------------------------------------------------
// Kernel 3: per-batch top-100 of 80000 (L2-hot), deterministic ties -> lowest
// index. Picked entries are marked -inf in scratch (regenerated each launch).
// Output: dets (B,100,5) floats, then labels (B,100) int32, concatenated.
// ---------------------------------------------------------------------------
__global__ __launch_bounds__(512) void topk_kernel(float* __restrict__ ws_scores,
                                                   const float* __restrict__ ws_boxes,
                                                   float* __restrict__ out)
{
  const int b   = blockIdx.x;
  const int tid = threadIdx.x;
  const int FL  = (NCLS - 1) * NPROP;              // 80000
  float*       sc = ws_scores + (size_t)b * FL;
  const float* bx = ws_boxes  + (size_t)b * FL * 4;
  float* dets   = out + (size_t)b * NDET * 5;
  int*   labels = (int*)(out + (size_t)BATCH * NDET * 5) + b * NDET;

  __shared__ float rv[512];
  __shared__ int   ri[512];

  for (int it = 0; it < NDET; ++it) {
    float best = NEG_INF; int bi = FL;
    for (int i = tid; i < FL; i += 512) {
      const float v = sc[i];
      if (v > best || (v == best && i < bi)) { best = v; bi = i; }
    }
    rv[tid] = best; ri[tid] = bi;
    __syncthreads();
    for (int s = 256; s > 0; s >>= 1) {
      if (tid < s) {
        const float v = rv[tid + s]; const int i2 = ri[tid + s];
        if (v > rv[tid] || (v == rv[tid] && i2 < ri[tid])) { rv[tid] = v; ri[tid] = i2; }
      }
      __syncthreads();
    }
    if (tid == 0) {
      const float v = rv[0]; const int i = ri[0];
      const bool valid = (v > NEG_INF) && (i < FL);
      const float4 bb = valid ? *(const float4*)(bx + (size_t)i * 4)
                              : make_float4(0, 0, 0, 0);
      dets[it * 5 + 0] = valid ? v : 0.0f;
      dets[it * 5 + 1] = bb.x; dets[it * 5 + 2] = bb.y;
      dets[it * 5 + 3] = bb.z; dets[it * 5 + 4] = bb.w;
      labels[it] = valid ? (i / NPROP + 1) : 0;
      if (i < FL) sc[i] = NEG_INF;                 // exclude from later rounds
    }
    __syncthreads();
  }
}

// ---------------------------------------------------------------------------
extern "C" void kernel_launch(void* const* d_in, const int* in_sizes, int n_in,
                              void* d_out, int out_size, void* d_ws, size_t ws_size,
                              hipStream_t stream)
{
  const float* cls  = (const float*)d_in[0];   // (2000, 81)
  const float* reg  = (const float*)d_in[1];   // (2000, 324)
  const float* prop = (const float*)d_in[2];   // (2000, 4)

  float* ws_scores = (float*)d_ws;                         // 160000 floats
  float* ws_boxes  = ws_scores + (size_t)NPAIR * NPROP;    // 640000 floats
  float* out = (float*)d_out;

  softmax_decode_kernel<<<16, 256, 0, stream>>>(cls, reg, prop, ws_scores, ws_boxes);
  nms_kernel<<<NPAIR, 256, 0, stream>>>(ws_scores, ws_boxes);
  topk_kernel<<<BATCH, 512, 0, stream>>>(ws_scores, ws_boxes, out);
}